// Loss_66992899883032
// MI455X (gfx1250) — compile-verified
//
#include <hip/hip_runtime.h>

typedef __attribute__((ext_vector_type(2))) float v2f;
typedef __attribute__((ext_vector_type(4))) float f4;
typedef __attribute__((ext_vector_type(8))) float v8f;

#define WH      65536u              // 256*256 pixels per plane
#define NPIX    (32u * WH)          // B * W * H = 2,097,152
#define GRID    1024
#define BLOCK   256

#define LN2f    0.69314718055994530942f

// Kernel 1: streaming Gaussian-likelihood partial sums.
// Each thread handles 2 chunks of 4 consecutive pixels (exact coverage, no tail).
__global__ __launch_bounds__(BLOCK) void gau_loss_partial(
    const float* __restrict__ x_mean,     // [B,3,WH]
    const float* __restrict__ x_var,      // [B,3,3,WH] (symmetric per pixel)
    const float* __restrict__ n_sigma,    // [B,1,1,WH] -> flat [NPIX]
    const float* __restrict__ target,     // [B,3,WH]
    float* __restrict__ partials)         // [GRID]
{
    const unsigned tid      = blockIdx.x * BLOCK + threadIdx.x;
    const unsigned nthreads = GRID * BLOCK;       // 262,144
    const unsigned nchunks  = NPIX / 4u;          // 524,288 -> exactly 2 iters/thread

    float acc = 0.0f;

    for (unsigned ci = tid; ci < nchunks; ci += nthreads) {
        const unsigned pix   = ci << 2;           // first pixel of float4 chunk
        const unsigned b     = pix >> 16;         // pix / WH
        const unsigned p     = pix & (WH - 1u);   // pix % WH (16B-aligned, never crosses plane)
        const unsigned mbase = b * (3u * WH) + p;
        const unsigned vbase = b * (9u * WH) + p;

        // 13 coalesced non-temporal 128-bit loads per chunk (read-once streams).
        f4 m0  = __builtin_nontemporal_load((const f4*)(x_mean + mbase));
        f4 m1  = __builtin_nontemporal_load((const f4*)(x_mean + mbase + WH));
        f4 m2  = __builtin_nontemporal_load((const f4*)(x_mean + mbase + 2u * WH));
        f4 t0  = __builtin_nontemporal_load((const f4*)(target + mbase));
        f4 t1  = __builtin_nontemporal_load((const f4*)(target + mbase + WH));
        f4 t2  = __builtin_nontemporal_load((const f4*)(target + mbase + 2u * WH));
        f4 v00 = __builtin_nontemporal_load((const f4*)(x_var + vbase));
        f4 v01 = __builtin_nontemporal_load((const f4*)(x_var + vbase + 1u * WH));
        f4 v02 = __builtin_nontemporal_load((const f4*)(x_var + vbase + 2u * WH));
        f4 v11 = __builtin_nontemporal_load((const f4*)(x_var + vbase + 4u * WH));
        f4 v12 = __builtin_nontemporal_load((const f4*)(x_var + vbase + 5u * WH));
        f4 v22 = __builtin_nontemporal_load((const f4*)(x_var + vbase + 8u * WH));
        f4 sg  = __builtin_nontemporal_load((const f4*)(n_sigma + pix));

#pragma unroll
        for (int j = 0; j < 4; ++j) {
            const float s2  = sg[j] * sg[j];
            const float a00 = v00[j] + s2, a01 = v01[j], a02 = v02[j];
            const float a11 = v11[j] + s2, a12 = v12[j], a22 = v22[j] + s2;
            // adjugate (symmetric) + determinant, closed form
            const float c00 = a11 * a22 - a12 * a12;
            const float c01 = a12 * a02 - a01 * a22;
            const float c02 = a01 * a12 - a11 * a02;
            const float c11 = a00 * a22 - a02 * a02;
            const float c12 = a01 * a02 - a00 * a12;
            const float c22 = a00 * a11 - a01 * a01;
            const float det = a00 * c00 + a01 * c01 + a02 * c02;
            const float d0 = t0[j] - m0[j];
            const float d1 = t1[j] - m1[j];
            const float d2 = t2[j] - m2[j];
            const float quad = d0 * d0 * c00 + d1 * d1 * c11 + d2 * d2 * c22 +
                               2.0f * (d0 * d1 * c01 + d0 * d2 * c02 + d1 * d2 * c12);
            // Single-instruction paths: v_rcp_f32 + v_log_f32 (det >= (0.1+s2)^3 > 0,
            // well-conditioned by construction, so approximate ops are safe).
            acc += quad * __builtin_amdgcn_rcpf(det) +
                   __builtin_amdgcn_logf(det) * LN2f;   // log2 -> ln
        }
    }

    // ---- wave reduction via V_WMMA_F32_16X16X4_F32 ----
    // A (16x4): lane L<16 holds A[L][0]=acc, A[L][1]=0; lane L>=16 holds A[L-16][2]=acc, A[L-16][3]=0.
    // B = all ones  =>  D[m][n] = acc[lane m] + acc[lane m+16].
    v2f amat; amat.x = acc;  amat.y = 0.0f;
    v2f bones; bones.x = 1.0f; bones.y = 1.0f;
    v8f cz = {};
    v8f d = __builtin_amdgcn_wmma_f32_16x16x4_f32(false, amat, false, bones,
                                                  (short)0, cz, false, false);
    // Sum the 8 D VGPRs: lanes 0-15 hold sum over rows 0-7, lanes 16-31 over rows 8-15.
    float half = d[0] + d[1] + d[2] + d[3] + d[4] + d[5] + d[6] + d[7];

    __shared__ float lds[2 * (BLOCK / 32)];
    const unsigned lane = threadIdx.x & 31u;
    const unsigned wave = threadIdx.x >> 5;
    if (lane == 0u)  lds[2u * wave]      = half;   // rows 0-7  half-sum
    if (lane == 16u) lds[2u * wave + 1u] = half;   // rows 8-15 half-sum
    __syncthreads();

    if (threadIdx.x == 0) {
        float s = 0.0f;
#pragma unroll
        for (int i = 0; i < 2 * (BLOCK / 32); ++i) s += lds[i];
        partials[blockIdx.x] = s;
    }
}

// Kernel 2: deterministic fixed-order reduction of GRID partials -> scalar loss.
__global__ __launch_bounds__(256) void gau_loss_finalize(
    const float* __restrict__ partials, float* __restrict__ out)
{
    __shared__ float sdata[256];
    float s = 0.0f;
    for (int i = threadIdx.x; i < GRID; i += 256) s += partials[i];
    sdata[threadIdx.x] = s;
    __syncthreads();
    for (int off = 128; off > 0; off >>= 1) {
        if ((int)threadIdx.x < off) sdata[threadIdx.x] += sdata[threadIdx.x + off];
        __syncthreads();
    }
    if (threadIdx.x == 0) {
        // loss = WEIGHT * mean((quad + logdet)/2) = sum / (2*NPIX)
        out[0] = sdata[0] * (0.5f / (float)NPIX);
    }
}

extern "C" void kernel_launch(void* const* d_in, const int* in_sizes, int n_in,
                              void* d_out, int out_size, void* d_ws, size_t ws_size,
                              hipStream_t stream) {
    const float* x_mean   = (const float*)d_in[0];
    const float* x_var    = (const float*)d_in[1];
    const float* n_sigma  = (const float*)d_in[2];
    const float* target   = (const float*)d_in[3];
    float* partials       = (float*)d_ws;          // needs GRID*4 = 4 KB of scratch
    float* out            = (float*)d_out;

    gau_loss_partial<<<GRID, BLOCK, 0, stream>>>(x_mean, x_var, n_sigma, target, partials);
    gau_loss_finalize<<<1, 256, 0, stream>>>(partials, out);
    (void)in_sizes; (void)n_in; (void)out_size; (void)ws_size;
}